// HermanModel_63917703299748
// MI455X (gfx1250) — compile-verified
//
#include <hip/hip_runtime.h>
#include <stdint.h>

// ---------------------------------------------------------------------------
// HermanModel spiking network, MI455X (gfx1250, wave32).
// Edge phase: async global->LDS (b128) double-buffered streaming of
//             (W, src, dst); random gather of act[src] (L2-resident);
//             FP32 global atomic scatter-add (L2 atomics).
// Neuron phase: JAX Threefry2x32 RNG + erfinv-based normal, spike + decay.
// Whole working set (~118 MB) is L2-resident (192 MB) after step 0.
// ---------------------------------------------------------------------------

#define AS1 __attribute__((address_space(1)))
#define AS3 __attribute__((address_space(3)))

typedef int v4i_t __attribute__((ext_vector_type(4)));

// Generic->AS pointer conversion via integer (LDS generic addr low 32 bits ==
// LDS byte offset on AMDGPU; global generic addr == global addr).
#define N_GLB(p)  ((AS1 int*)(unsigned long long)(const void*)(p))
#define N_LDS(p)  ((AS3 int*)(unsigned int)(unsigned long long)(const void*)(p))
#define N_GLB4(p) ((AS1 v4i_t*)(unsigned long long)(const void*)(p))
#define N_LDS4(p) ((AS3 v4i_t*)(unsigned int)(unsigned long long)(const void*)(p))

#if defined(__HIP_DEVICE_COMPILE__)
  #if __has_builtin(__builtin_amdgcn_global_load_async_to_lds_b32)
    #define ASYNC_LOAD_B32(g, l) \
      __builtin_amdgcn_global_load_async_to_lds_b32(N_GLB(g), N_LDS(l), 0, 0)
  #else
    #define ASYNC_LOAD_B32(g, l)                                              \
      asm volatile("global_load_async_to_lds_b32 %0, %1, off"                 \
                   :: "v"((unsigned int)(unsigned long long)(const void*)(l)),\
                      "v"((const void*)(g)) : "memory")
  #endif
  #if __has_builtin(__builtin_amdgcn_global_load_async_to_lds_b128)
    #define ASYNC_LOAD_B128(g, l) \
      __builtin_amdgcn_global_load_async_to_lds_b128(N_GLB4(g), N_LDS4(l), 0, 0)
    #define TILE_ASYNC_OPS 3
  #else
    #define ASYNC_LOAD_B128(g, l) do {                                        \
      const int* _g = (const int*)(const void*)(g);                           \
      int*       _l = (int*)(void*)(l);                                       \
      ASYNC_LOAD_B32(_g + 0, _l + 0); ASYNC_LOAD_B32(_g + 1, _l + 1);         \
      ASYNC_LOAD_B32(_g + 2, _l + 2); ASYNC_LOAD_B32(_g + 3, _l + 3);         \
    } while (0)
    #define TILE_ASYNC_OPS 12
  #endif
  #if __has_builtin(__builtin_amdgcn_s_wait_asynccnt)
    #define ASYNC_WAIT(n) __builtin_amdgcn_s_wait_asynccnt(n)
  #else
    #define ASYNC_WAIT(n) asm volatile("s_wait_asynccnt %0" :: "n"(n))
  #endif
#else
  #define ASYNC_LOAD_B32(g, l)  ((void)0)
  #define ASYNC_LOAD_B128(g, l) ((void)0)
  #define ASYNC_WAIT(n)         ((void)0)
  #define TILE_ASYNC_OPS 3
#endif

// ---------------- Threefry2x32 (exact JAX schedule) ----------------
__host__ __device__ __forceinline__ unsigned rotl32(unsigned x, int r) {
  return (x << r) | (x >> (32 - r));
}

__host__ __device__ __forceinline__ void threefry2x32(
    unsigned k0, unsigned k1, unsigned c0, unsigned c1,
    unsigned& o0, unsigned& o1) {
  unsigned ks0 = k0, ks1 = k1, ks2 = k0 ^ k1 ^ 0x1BD11BDAu;
  unsigned x0 = c0 + ks0, x1 = c1 + ks1;
#define TF_R(r) { x0 += x1; x1 = rotl32(x1, r); x1 ^= x0; }
  TF_R(13) TF_R(15) TF_R(26) TF_R(6)   x0 += ks1; x1 += ks2 + 1u;
  TF_R(17) TF_R(29) TF_R(16) TF_R(24)  x0 += ks2; x1 += ks0 + 2u;
  TF_R(13) TF_R(15) TF_R(26) TF_R(6)   x0 += ks0; x1 += ks1 + 3u;
  TF_R(17) TF_R(29) TF_R(16) TF_R(24)  x0 += ks1; x1 += ks2 + 4u;
  TF_R(13) TF_R(15) TF_R(26) TF_R(6)   x0 += ks2; x1 += ks0 + 5u;
#undef TF_R
  o0 = x0; o1 = x1;
}

// ---------------- erfinv (Giles 2010, single precision) ----------------
__device__ __forceinline__ float erfinv_f32(float x) {
  float w = -__logf((1.0f - x) * (1.0f + x));
  float p;
  if (w < 5.0f) {
    w -= 2.5f;
    p =  2.81022636e-08f;
    p =  3.43273939e-07f + p * w;
    p = -3.52338770e-06f + p * w;
    p = -4.39150654e-06f + p * w;
    p =  2.18580870e-04f + p * w;
    p = -1.25372503e-03f + p * w;
    p = -4.17768164e-03f + p * w;
    p =  2.46640727e-01f + p * w;
    p =  1.50140941e+00f + p * w;
  } else {
    w = sqrtf(w) - 3.0f;
    p = -2.00214257e-04f;
    p =  1.00950558e-04f + p * w;
    p =  1.34934322e-03f + p * w;
    p = -3.67342844e-03f + p * w;
    p =  5.73950773e-03f + p * w;
    p = -7.62246130e-03f + p * w;
    p =  9.43887047e-03f + p * w;
    p =  1.00167406e+00f + p * w;
    p =  2.83297682e+00f + p * w;
  }
  return p * x;
}

// bits -> uniform in [nextafter(-1,0), 1) exactly like jax._uniform
__device__ __forceinline__ float bits_to_pm1(unsigned b) {
  float u01 = __uint_as_float((b >> 9) | 0x3f800000u) - 1.0f;   // [0,1)
  const float lo = -0.99999994f;                                 // nextafter(-1,0)
  float v = u01 * (1.0f - lo) + lo;
  return fmaxf(lo, v);
}

// ---------------- Kernels ----------------
__global__ void __launch_bounds__(256) herman_init_kernel(
    const float* __restrict__ act_in, float* __restrict__ act,
    float* __restrict__ agg, int n) {
  int i = blockIdx.x * 256 + threadIdx.x;
  if (i < n) { act[i] = act_in[i]; agg[i] = 0.0f; }
}

#define EDGE_TILE 1024   // 256 threads x 4 edges, b128 per array per thread

// Edge aggregation: async-to-LDS (b128) double-buffered streaming of W/src/dst.
// Each thread owns its own 16B LDS slots per array (no cross-thread sharing ->
// only s_wait_asynccnt needed, no barriers). Full 1024-edge tiles; tail via
// plain loads.
__global__ void __launch_bounds__(256) herman_edge_kernel(
    const float* __restrict__ W, const int* __restrict__ src,
    const int* __restrict__ dst, const float* __restrict__ act,
    float* __restrict__ agg, int nEdges) {
  __shared__ int   sSrc[2][EDGE_TILE];
  __shared__ int   sDst[2][EDGE_TILE];
  __shared__ float sW  [2][EDGE_TILE];

  const int tid    = threadIdx.x;
  const int lane4  = tid << 2;              // this thread's 4-edge slot
  const int nTiles = nEdges / EDGE_TILE;    // full tiles
  const int stride = gridDim.x;

  int tile = blockIdx.x;
  if (tile < nTiles) {
    int buf = 0;
    {   // prefetch first tile into buffer 0 (3 b128 async ops per thread)
      int e = tile * EDGE_TILE + lane4;
      ASYNC_LOAD_B128(src + e, &sSrc[0][lane4]);
      ASYNC_LOAD_B128(dst + e, &sDst[0][lane4]);
      ASYNC_LOAD_B128(W   + e, &sW  [0][lane4]);
    }
    for (; tile < nTiles; tile += stride) {
      int nextTile = tile + stride;
      if (nextTile < nTiles) {
        int e = nextTile * EDGE_TILE + lane4;   // all 256 lanes active
        ASYNC_LOAD_B128(src + e, &sSrc[buf ^ 1][lane4]);
        ASYNC_LOAD_B128(dst + e, &sDst[buf ^ 1][lane4]);
        ASYNC_LOAD_B128(W   + e, &sW  [buf ^ 1][lane4]);
        ASYNC_WAIT(TILE_ASYNC_OPS);         // current tile's ops complete
      } else {
        ASYNC_WAIT(0);
      }
      int4   s4 = *(const int4*)  &sSrc[buf][lane4];   // ds_load_b128
      int4   d4 = *(const int4*)  &sDst[buf][lane4];
      float4 w4 = *(const float4*)&sW  [buf][lane4];
      unsafeAtomicAdd(&agg[d4.x], w4.x * act[s4.x]);   // L2 gather + L2 atomic
      unsafeAtomicAdd(&agg[d4.y], w4.y * act[s4.y]);
      unsafeAtomicAdd(&agg[d4.z], w4.z * act[s4.z]);
      unsafeAtomicAdd(&agg[d4.w], w4.w * act[s4.w]);
      buf ^= 1;
    }
  }
  // tail edges (nEdges not a multiple of EDGE_TILE) — plain path
  for (int e = nTiles * EDGE_TILE + blockIdx.x * 256 + tid; e < nEdges;
       e += gridDim.x * 256) {
    unsafeAtomicAdd(&agg[dst[e]], W[e] * act[src[e]]);
  }
}

// Neuron update: JAX-exact RNG layout. random_bits(key,(n,)) pairs counters
// (i, i+n/2); lane-0 output for i < n/2, lane-1 output otherwise.
__global__ void __launch_bounds__(256) herman_update_kernel(
    float* __restrict__ act, float* __restrict__ agg, float* __restrict__ out,
    int n, int nSteps, int t,
    unsigned kn0, unsigned kn1, unsigned kf0, unsigned kf1) {
  int i = blockIdx.x * 256 + threadIdx.x;
  if (i >= n) return;
  const int h   = n >> 1;                   // n is even (100000)
  const bool hi = (i >= h);
  unsigned c0 = (unsigned)(hi ? (i - h) : i);
  unsigned c1 = c0 + (unsigned)h;

  unsigned o0, o1;
  threefry2x32(kn0, kn1, c0, c1, o0, o1);
  unsigned bn = hi ? o1 : o0;
  threefry2x32(kf0, kf1, c0, c1, o0, o1);
  unsigned bf = hi ? o1 : o0;

  const float SQRT2 = 1.41421356237f;
  float noise = 0.3f * (SQRT2 * erfinv_f32(bits_to_pm1(bn)));
  float fnorm =         SQRT2 * erfinv_f32(bits_to_pm1(bf));
  float filt  = (fnorm > 1.5f) ? 1.0f : 0.0f;

  float a = agg[i];
  float l = 0.025f * a + 0.001f * (1.0f + noise * filt);
  float spike = (l > 0.001378f) ? 1.0f : 0.0f;

  out[(size_t)i * (size_t)nSteps + (size_t)t] = spike;
  act[i] = a + spike - a * 0.01f;           // agg + spikes - agg/tau*dt
  agg[i] = 0.0f;                            // reset accumulator for next step
}

// ---------------- Host side ----------------
extern "C" void kernel_launch(void* const* d_in, const int* in_sizes, int n_in,
                              void* d_out, int out_size, void* d_ws, size_t ws_size,
                              hipStream_t stream) {
  const float* act_in = (const float*)d_in[0];
  const float* W      = (const float*)d_in[1];
  const int*   ei     = (const int*)d_in[2];

  const int n = in_sizes[0];                // 100000 neurons
  const int E = in_sizes[1];                // 6.4M edges
  const int* src = ei;                      // edge_index[0]
  const int* dst = ei + E;                  // edge_index[1]
  const int nSteps = (n > 0) ? (out_size / n) : 0;

  float* out = (float*)d_out;
  float* act = (float*)d_ws;                // N floats
  float* agg = act + n;                     // N floats

  const int ublocks = (n + 255) / 256;
  herman_init_kernel<<<ublocks, 256, 0, stream>>>(act_in, act, agg, n);

  int nTiles  = E / EDGE_TILE;
  int eblocks = nTiles < 2048 ? (nTiles > 0 ? nTiles : 1) : 2048;

  for (int t = 0; t < nSteps; ++t) {
    // k = fold_in(key(42), t) = threefry((0,42), (0,t))
    unsigned k0, k1;
    threefry2x32(0u, 42u, 0u, (unsigned)t, k0, k1);
    // split(k): counts iota(4)=[0,1,2,3] -> halves x0=[0,1], x1=[2,3]
    //   lane0 pair (0,2) -> (a0,b0); lane1 pair (1,3) -> (a1,b1)
    //   kn = (a0,a1) [lane-0 outputs], kf = (b0,b1) [lane-1 outputs]
    unsigned a0, b0, a1, b1;
    threefry2x32(k0, k1, 0u, 2u, a0, b0);
    threefry2x32(k0, k1, 1u, 3u, a1, b1);

    herman_edge_kernel<<<eblocks, 256, 0, stream>>>(W, src, dst, act, agg, E);
    herman_update_kernel<<<ublocks, 256, 0, stream>>>(
        act, agg, out, n, nSteps, t, a0, a1, b0, b1);
  }
}